// DigitCapsuleLayer_31619549233684
// MI455X (gfx1250) — compile-verified
//
#include <hip/hip_runtime.h>
#include <math.h>

typedef __attribute__((ext_vector_type(2))) float v2f;
typedef __attribute__((ext_vector_type(8))) float v8f;

// Problem constants (from reference):
#define BB 128      // batch
#define RR 2048     // routes
#define CC 32       // capsules
#define II 16       // in-cap dim
#define OO 32       // out-cap dim
#define KK (RR*II)  // 32768  GEMM K
#define NN (CC*OO)  // 1024   GEMM N

// One block = 8 waves (wave32), 256 threads.
// Block (mi, c): 16 batches (m_base = mi*16) x one capsule c (n = c*32 .. c*32+31).
// Each wave: full 16x32 output tile, 1/8 of K (256 routes, stride-8 interleaved).
__global__ __launch_bounds__(256)
void caps_fused_kernel(const float* __restrict__ x,   // [B, R, I]  = [128, 32768] row-major
                       const float* __restrict__ W,   // [R, C, O, I]
                       float* __restrict__ out)       // [B, C, O, 1]
{
    __shared__ float lds[8 * 512];   // 8 wave-partials of 16x32 floats = 16 KB

    const int tid  = threadIdx.x;
    const int wv   = tid >> 5;        // wave in block, 0..7
    const int lane = tid & 31;
    const int lrow = lane & 15;       // M index (A/C) or N index (B/C)
    const int lhi  = lane >> 4;       // 0/1: selects K pair within WMMA operand

    const int mi     = blockIdx.x & 7;   // 8 m-blocks
    const int c      = blockIdx.x >> 3;  // 32 capsules
    const int m_base = mi * 16;

    // A operand base: x[(m_base+lrow)*KK + r*16 + 4t + 2*lhi], pair of 2 floats.
    const float* xp = x + (size_t)(m_base + lrow) * KK + 2 * lhi;
    // B operand base: W[r*16384 + c*512 + (h*16+lrow)*16 + 4t + 2*lhi], pair of 2 floats.
    const float* wp = W + (size_t)c * 512 + (size_t)lrow * 16 + 2 * lhi;

    v8f acc0 = {};   // o = 0..15  half
    v8f acc1 = {};   // o = 16..31 half

    for (int it = 0; it < RR / 8; ++it) {
        const int r = it * 8 + wv;                    // interleaved K split over waves
        const float* xr = xp + r * II;                // this route's x pairs
        const float* wr = wp + (size_t)r * (CC*OO*II);// this route's W block (contig 2KB per c)

        if (it < RR / 8 - 1) {
            // speculative prefetch of next iteration's W block (global_prefetch_b8)
            __builtin_prefetch((const void*)(wr + (size_t)8 * (CC*OO*II)), 0, 1);
        }

        #pragma unroll
        for (int t = 0; t < 4; ++t) {                 // 4 WMMA K-steps cover i = 0..15
            v2f a  = *(const v2f*)(xr + 4 * t);
            v2f b0 = *(const v2f*)(wr + 4 * t);           // o half 0
            v2f b1 = *(const v2f*)(wr + 256 + 4 * t);     // o half 1 (+256 floats)
            acc0 = __builtin_amdgcn_wmma_f32_16x16x4_f32(
                       false, a, false, b0, (short)0, acc0, false, false);
            acc1 = __builtin_amdgcn_wmma_f32_16x16x4_f32(
                       false, a, false, b1, (short)0, acc1, false, false);
        }
    }

    // ---- write per-wave partial tile to LDS ----
    // C/D layout: VGPR j, lanes 0-15 -> (m=j, n=lane), lanes 16-31 -> (m=j+8, n=lane-16)
    {
        float* lw = lds + wv * 512;
        #pragma unroll
        for (int j = 0; j < 8; ++j) {
            const int m = j + 8 * lhi;
            lw[m * 32 + lrow]      = acc0[j];
            lw[m * 32 + 16 + lrow] = acc1[j];
        }
    }
    __syncthreads();

    // ---- reduce 8 wave partials, scale by 1/R (uniform routing coefficients) ----
    #pragma unroll
    for (int half = 0; half < 2; ++half) {
        const int p = tid + half * 256;     // position 0..511, each owned by one thread
        float s = 0.0f;
        #pragma unroll
        for (int w = 0; w < 8; ++w) s += lds[w * 512 + p];
        lds[p] = s * (1.0f / (float)RR);
    }
    __syncthreads();

    // ---- squash over o (tile holds all 32 o's of capsule c) and store ----
    if (tid < 16) {
        float sq = 0.0f;
        #pragma unroll
        for (int o = 0; o < 32; ++o) {
            const float v = lds[tid * 32 + o];
            sq += v * v;
        }
        const float scale = sq / ((1.0f + sq) * sqrtf(sq));
        float* op = out + (size_t)(m_base + tid) * NN + c * 32;
        #pragma unroll
        for (int o = 0; o < 32; ++o)
            op[o] = lds[tid * 32 + o] * scale;
    }
}

extern "C" void kernel_launch(void* const* d_in, const int* in_sizes, int n_in,
                              void* d_out, int out_size, void* d_ws, size_t ws_size,
                              hipStream_t stream) {
    const float* x = (const float*)d_in[0];   // [128, 2048, 16]
    const float* W = (const float*)d_in[1];   // [1, 2048, 32, 32, 16]
    float* out = (float*)d_out;               // [128, 32, 32, 1]

    dim3 grid(256);   // 8 m-blocks * 32 capsules
    dim3 block(256);  // 8 waves
    hipLaunchKernelGGL(caps_fused_kernel, grid, block, 0, stream, x, W, out);
}